// BallQLossSeq_50775103373767
// MI455X (gfx1250) — compile-verified
//
#include <hip/hip_runtime.h>
#include <hip/hip_bf16.h>
#include <math.h>

// CDNA5 / gfx1250, wave32. Ball-query (first-16 in ascending order, radius 1)
// via V_WMMA_F32_16X16X4_F32 Gram-matrix tiles + ballot/prefix-popcount ordered
// append, fused with the flow-distance loss and a deterministic reduction.
// Round 2: software-pipelined B-tile loads (prefetch next tile before using
// the current one) so the per-tile global load overlaps the WMMA + selection
// VALU work instead of stalling on s_wait_loadcnt 0 every iteration.

typedef float v2f __attribute__((ext_vector_type(2)));
typedef float v8f __attribute__((ext_vector_type(8)));

#define KNN 16
#define RAD2 1.0f
#define QB 128            // queries per block
#define TPB 256           // 8 wave32 waves
#define ROWS_PER_WAVE 16

__global__ __launch_bounds__(TPB) void ballq_loss_kernel(
    const float* __restrict__ pc,     // [N,3]
    const float* __restrict__ flow,   // [SEQ,N,3]
    float* __restrict__ partials,     // [gridDim.x]
    int N, int SEQ)
{
  __shared__ int   idxbuf[QB][KNN];   // neighbor lists for this block's queries
  __shared__ int   cnt_lds[QB];
  __shared__ float wsum[TPB / 32];

  const int tid  = threadIdx.x;
  const int lane = tid & 31;
  const int wave = tid >> 5;
  const int hi   = lane >> 4;          // 0: lanes 0-15, 1: lanes 16-31
  const int n16  = lane & 15;
  const int qblk = blockIdx.x * QB;                 // first query of block
  const int qw   = qblk + wave * ROWS_PER_WAVE;     // first query row of wave

  // ---- A matrix (16x4 f32, MxK): lane holds query M = lane&15.
  // Layout: VGPR0 = K0 (lanes 0-15) / K2 (lanes 16-31); VGPR1 = K1 / K3.
  // Coordinates (x,y,z,0) padded to K=4.
  const int qm = qw + n16;
  float qx = pc[3*qm+0], qy = pc[3*qm+1], qz = pc[3*qm+2];
  v2f a;
  a.x = hi ? qz : qx;
  a.y = hi ? 0.0f : qy;

  // Per-lane |q|^2 for the rows this half-wave sees in accumulator VGPR r:
  // row = r + 8*hi (matches the 16x16 f32 C/D layout).
  float sqq[8];
#pragma unroll
  for (int r = 0; r < 8; ++r) {
    int qi = qw + r + 8*hi;
    float x = pc[3*qi+0], y = pc[3*qi+1], z = pc[3*qi+2];
    sqq[r] = x*x + y*y + z*z;
  }

  int cnt[8];
#pragma unroll
  for (int r = 0; r < 8; ++r) cnt[r] = 0;

  const int ntiles = N >> 4;

  // Prime the pipeline: load B tile 0.
  float nsx = pc[3*n16+0], nsy = pc[3*n16+1], nsz = pc[3*n16+2];

  for (int jt = 0; jt < ntiles; ++jt) {
    const int jb = jt << 4;

    // Consume the prefetched tile.
    float sx = nsx, sy = nsy, sz = nsz;

    // Branchless prefetch of tile jt+1 (clamped to tile 0 on the last iter;
    // result unused there). Issued before any use of sx/sy/sz so the load
    // overlaps this tile's WMMA + selection work.
    {
      int jn = (jt + 1 < ntiles) ? (jb + 16 + n16) : n16;
      nsx = pc[3*jn+0]; nsy = pc[3*jn+1]; nsz = pc[3*jn+2];
    }

    float sqs = sx*sx + sy*sy + sz*sz;     // |s_N|^2, same for every r

    // ---- B matrix (4x16 f32, KxN): lane holds source column N = lane&15.
    // VGPR0 = K0 row (lanes 0-15) / K2 row (lanes 16-31); VGPR1 = K1 / K3.
    v2f b;
    b.x = hi ? sz : sx;
    b.y = hi ? 0.0f : sy;

    // One WMMA = 16x16 tile of dot(q_M, s_N), K padded 3->4 with zeros.
    v8f c = {};
    c = __builtin_amdgcn_wmma_f32_16x16x4_f32(
        /*neg_a=*/false, a, /*neg_b=*/false, b,
        /*c_mod=*/(short)0, c, /*reuse_a=*/false, /*reuse_b=*/false);

#pragma unroll
    for (int r = 0; r < 8; ++r) {
      // element (M = r + 8*hi, N = lane&15)
      float d2 = sqq[r] + sqs - 2.0f * c[r];
      unsigned m   = __builtin_amdgcn_ballot_w32(d2 < RAD2);
      unsigned m16 = (m >> (hi << 4)) & 0xffffu;   // this half's row mask
      int cr  = cnt[r];
      int pre = __builtin_popcount(m16 & ((1u << n16) - 1u));
      int pos = cr + pre;                          // ordered insert slot
      if (((m16 >> n16) & 1u) && pos < KNN) {
        idxbuf[wave * ROWS_PER_WAVE + r + 8*hi][pos] = jb + n16;
      }
      cr += __builtin_popcount(m16);
      cnt[r] = cr > KNN ? KNN : cr;
    }

    // Wave-uniform early exit once all 16 rows have KNN hits.
    bool notdone = false;
#pragma unroll
    for (int r = 0; r < 8; ++r) notdone |= (cnt[r] < KNN);
    if (__builtin_amdgcn_ballot_w32(notdone) == 0u) break;
  }

  // Publish per-row counts (replicated across the half-wave; one lane writes).
  if (n16 == 0) {
#pragma unroll
    for (int r = 0; r < 8; ++r)
      cnt_lds[wave * ROWS_PER_WAVE + r + 8*hi] = cnt[r];
  }
  __syncthreads();

  // Pad short rows with the first hit (self-hit guarantees cnt >= 1).
  if (tid < QB) {
    int c0 = cnt_lds[tid];
    int f  = idxbuf[tid][0];
    for (int p = c0; p < KNN; ++p) idxbuf[tid][p] = f;
  }
  __syncthreads();

  // ---- Phase 2: loss over this block's (i,k) pairs, all SEQ sequences ----
  float acc = 0.0f;
  const int* idxflat = &idxbuf[0][0];
  for (int p = tid; p < QB * KNN; p += TPB) {
    int i = qblk + (p >> 4);
    int j = idxflat[p];
    for (int s = 0; s < SEQ; ++s) {
      const float* fi = flow + ((size_t)s * N + i) * 3;
      const float* fj = flow + ((size_t)s * N + j) * 3;
      float dx = fi[0] - fj[0];
      float dy = fi[1] - fj[1];
      float dz = fi[2] - fj[2];
      float d2v = dx*dx + dy*dy + dz*dz;
      acc += sqrtf(fmaxf(d2v, 1e-24f));
    }
  }

  // Deterministic block reduction (fixed tree order, no float atomics).
#pragma unroll
  for (int off = 16; off >= 1; off >>= 1)
    acc += __shfl_xor(acc, off, 32);
  if (lane == 0) wsum[wave] = acc;
  __syncthreads();
  if (tid == 0) {
    float s = 0.0f;
    for (int wv = 0; wv < TPB / 32; ++wv) s += wsum[wv];
    partials[blockIdx.x] = s;
  }
}

__global__ void ballq_finalize_kernel(const float* __restrict__ partials,
                                      float* __restrict__ out,
                                      int nblocks, float scale)
{
  if (threadIdx.x == 0) {
    float s = 0.0f;
    for (int i = 0; i < nblocks; ++i) s += partials[i];   // fixed order
    out[0] = s * scale;
  }
}

extern "C" void kernel_launch(void* const* d_in, const int* in_sizes, int n_in,
                              void* d_out, int out_size, void* d_ws, size_t ws_size,
                              hipStream_t stream) {
  const float* pc   = (const float*)d_in[0];   // [1, N, 3] f32
  const float* flow = (const float*)d_in[1];   // [SEQ, N, 3] f32
  float* out        = (float*)d_out;           // scalar f32
  float* partials   = (float*)d_ws;            // nblocks floats

  const int N   = in_sizes[0] / 3;             // 8192
  const int SEQ = in_sizes[1] / (3 * N);       // 4
  const int nblocks = N / QB;                  // 64

  ballq_loss_kernel<<<nblocks, TPB, 0, stream>>>(pc, flow, partials, N, SEQ);

  const float scale = 1.0f / ((float)SEQ * (float)N * (float)KNN);
  ballq_finalize_kernel<<<1, 32, 0, stream>>>(partials, out, nblocks, scale);
}